// T2SBlock_11759620456876
// MI455X (gfx1250) — compile-verified
//
#include <hip/hip_runtime.h>
#include <hip/hip_bf16.h>

// Problem constants (from reference)
#define Bn   32
#define KVN  2048
#define Hn   1024
#define NHn  16
#define DHn  64
#define FFn  4096

// TDM tiling for the KV stream
#define CHUNK 64            // kv rows per TDM tile
#define NCH   (KVN / CHUNK) // 32 chunks
#define KROW  65            // LDS row stride in floats (64 + 1 pad via TDM pad_enable)

typedef __attribute__((ext_vector_type(2))) float v2f;
typedef __attribute__((ext_vector_type(8))) float v8f;
typedef __attribute__((ext_vector_type(4))) unsigned int u32x4;
typedef __attribute__((ext_vector_type(8))) int i32x8;
typedef __attribute__((ext_vector_type(4))) int i32x4;

// ---------------------------------------------------------------------------
// TDM: async 2-D tile load (rows x 64 floats, global stride H) into LDS with
// 1-DWORD padding after every 64 DWORDs (-> 65-float LDS row stride).
// ---------------------------------------------------------------------------
__device__ __forceinline__ i32x8 tdm_group1_kv()
{
    i32x8 g1 = {};
    // w0: workgroup_mask=0 | data_size=2 (4B) [17:16] | pad_enable [20]
    //     | pad_interval=5 (64 DWORDs) [24:22] | pad_amount=0 (1 DWORD) [31:25]
    g1[0] = (2 << 16) | (1 << 20) | (5 << 22);
    // tensor_dim0 [79:48] = 64 : w1[31:16] low half, w2[15:0] high half
    g1[1] = (DHn & 0xffff) << 16;
    // tensor_dim1 [111:80] = KVN : w2[31:16] low half, w3[15:0] high half
    g1[2] = (KVN & 0xffff) << 16;
    // tile_dim0 [127:112] = 64
    g1[3] = (DHn & 0xffff) << 16;
    // w4: tile_dim1 [143:128] = CHUNK rows, tile_dim2 [159:144] = 0
    g1[4] = CHUNK;
    // tensor_dim0_stride [207:160] = Hn : w5 low32, w6[15:0] high16
    g1[5] = Hn;
    g1[6] = 0;   // + tensor_dim1_stride low16 (unused, 2-D)
    g1[7] = 0;   // tensor_dim1_stride high32 (unused, 2-D)
    return g1;
}

__device__ __forceinline__ void tdm_load_2d(unsigned long long ga, unsigned ldsoff,
                                            i32x8 g1)
{
    u32x4 g0;
    g0[0] = 1u;                                   // count=1, user descriptor
    g0[1] = ldsoff;                               // lds_addr (bytes)
    g0[2] = (unsigned)(ga & 0xffffffffull);       // global_addr[31:0]
    g0[3] = (unsigned)((ga >> 32) & 0x01ffffffull)// global_addr[56:32]
          | 0x80000000u;                          // type=2 ("image") [127:126]
    i32x4 z4 = {};
    i32x8 z8 = {};
    __builtin_amdgcn_tensor_load_to_lds(g0, g1, z4, z4, z8, 0);
}

// ---------------------------------------------------------------------------
// WMMA GEMM:  C[M,N] = A[M,K] @ W[N,K]^T + bias, optional ReLU.
// One wave (32 threads) per 16x16 output tile; K-loop of 16x16x4 fp32 WMMAs.
// ---------------------------------------------------------------------------
__global__ void __launch_bounds__(32)
wmma_gemm_xWT(const float* __restrict__ A,
              const float* __restrict__ W,
              const float* __restrict__ bias,
              float* __restrict__ C,
              int M, int N, int K, int relu)
{
    const int tiles_n = N >> 4;
    const int mt = blockIdx.x / tiles_n;
    const int nt = blockIdx.x % tiles_n;
    const int lane = threadIdx.x;       // 0..31
    const int half = lane >> 4;         // 0: K pair {0,1}, 1: K pair {2,3}
    const int l16  = lane & 15;

    const float* __restrict__ ap = A + (size_t)(mt * 16 + l16) * K; // A row
    const float* __restrict__ bp = W + (size_t)(nt * 16 + l16) * K; // B col (= W row)

    v8f acc = {};
    for (int k0 = 0; k0 < K; k0 += 4) {
        const int ko = k0 + half * 2;
        v2f a, b;
        a.x = ap[ko]; a.y = ap[ko + 1];
        b.x = bp[ko]; b.y = bp[ko + 1];
        acc = __builtin_amdgcn_wmma_f32_16x16x4_f32(
            false, a, false, b, (short)0, acc, false, false);
    }

    const int col = nt * 16 + l16;
    const float bv = bias[col];
#pragma unroll
    for (int r = 0; r < 8; ++r) {
        const int row = mt * 16 + half * 8 + r;
        float v = acc[r] + bv;
        if (relu) v = fmaxf(v, 0.0f);
        C[(size_t)row * N + col] = v;
    }
}

// ---------------------------------------------------------------------------
// Copy ik/iv slices of fused qkv into the output tuple slots.
// ---------------------------------------------------------------------------
__global__ void split_qkv_kernel(const float* __restrict__ qkv,
                                 float* __restrict__ ik,
                                 float* __restrict__ iv)
{
    const int i = blockIdx.x * blockDim.x + threadIdx.x;   // 0 .. B*H-1
    const int b = i >> 10;
    const int c = i & (Hn - 1);
    ik[i] = qkv[b * 3 * Hn + Hn + c];
    iv[i] = qkv[b * 3 * Hn + 2 * Hn + c];
}

// ---------------------------------------------------------------------------
// Attention (q_len=1), one workgroup per (batch, head), 256 threads.
// K and V streamed through LDS by the Tensor Data Mover, double-buffered,
// paced with TENSORcnt.  TDM pad_enable gives a 65-float LDS row stride so
// all LDS reads below are bank-conflict-free.
// ---------------------------------------------------------------------------
__global__ void __launch_bounds__(256)
attn_kernel(const float* __restrict__ qkv,      // [B, 3H]
            const float* __restrict__ k_cache,  // [B, KV, H]
            const float* __restrict__ v_cache,  // [B, KV, H]
            float* __restrict__ out)            // [B, H]
{
    __shared__ float buf[2][CHUNK * KROW];   // 2 x 16.6 KB TDM tiles
    __shared__ float sc[KVN + 16];           // scores (KV+1 used)
    __shared__ float red[256];
    __shared__ float qs[DHn];

    const int b    = blockIdx.x / NHn;
    const int h    = blockIdx.x % NHn;
    const int tid  = threadIdx.x;
    const int wave = tid >> 5;
    const float scale = 0.125f;              // 1/sqrt(64)

    const unsigned long long kbase = (unsigned long long)(size_t)k_cache
        + ((size_t)b * KVN * Hn + (size_t)h * DHn) * 4ull;
    const unsigned long long vbase = (unsigned long long)(size_t)v_cache
        + ((size_t)b * KVN * Hn + (size_t)h * DHn) * 4ull;
    const unsigned lds_buf0 = (unsigned)(size_t)&buf[0][0]; // low 32 bits = LDS addr
    const unsigned lds_buf1 = (unsigned)(size_t)&buf[1][0];
    const i32x8 g1 = tdm_group1_kv();
    const unsigned long long chunk_bytes = (unsigned long long)CHUNK * Hn * 4ull;

    const float* __restrict__ qp   = qkv + (size_t)b * 3 * Hn + h * DHn;
    const float* __restrict__ knew = qkv + (size_t)b * 3 * Hn + Hn + h * DHn;
    const float* __restrict__ vnew = qkv + (size_t)b * 3 * Hn + 2 * Hn + h * DHn;

    if (tid < DHn) qs[tid] = qp[tid];

    // ---- preload K chunk 0 ----
    if (wave == 0) tdm_load_2d(kbase, lds_buf0, g1);

    // ---- phase 1: scores over KV cache, TDM double-buffered ----
    const int qrt = tid >> 6;        // quarter 0..3 (16 d-elements each)
    const int row = tid & 63;        // row within chunk
    float lmax = -3.4e38f;
    for (int c = 0; c < NCH; ++c) {
        if (wave == 0) {
            if (c + 1 < NCH) {
                tdm_load_2d(kbase + (unsigned long long)(c + 1) * chunk_bytes,
                            ((c + 1) & 1) ? lds_buf1 : lds_buf0, g1);
                __builtin_amdgcn_s_wait_tensorcnt(1);
            } else {
                __builtin_amdgcn_s_wait_tensorcnt(0);
            }
        }
        __syncthreads();
        // 4 threads per row, 16 d-elements each
        const float* __restrict__ kr = &buf[c & 1][row * KROW + qrt * 16];
        float s = 0.0f;
#pragma unroll
        for (int j = 0; j < 16; ++j) s += qs[qrt * 16 + j] * kr[j];
        red[tid] = s;
        __syncthreads();
        if (tid < CHUNK) {
            const float sv = (red[tid] + red[tid + 64] + red[tid + 128]
                            + red[tid + 192]) * scale;
            sc[c * CHUNK + tid] = sv;
            lmax = fmaxf(lmax, sv);
        }
        __syncthreads();
    }

    // appended K row (from this step's projection)
    if (tid == 0) {
        float s = 0.0f;
#pragma unroll
        for (int d = 0; d < DHn; ++d) s += qs[d] * knew[d];
        s *= scale;
        sc[KVN] = s;
        lmax = fmaxf(lmax, s);
    }

    // ---- softmax ----
    red[tid] = lmax;
    __syncthreads();
    for (int s2 = 128; s2 > 0; s2 >>= 1) {
        if (tid < s2) red[tid] = fmaxf(red[tid], red[tid + s2]);
        __syncthreads();
    }
    const float bmax = red[0];
    __syncthreads();

    float lsum = 0.0f;
    for (int k = tid; k < KVN + 1; k += 256) {
        const float e = __expf(sc[k] - bmax);
        sc[k] = e;
        lsum += e;
    }
    red[tid] = lsum;
    __syncthreads();
    for (int s2 = 128; s2 > 0; s2 >>= 1) {
        if (tid < s2) red[tid] += red[tid + s2];
        __syncthreads();
    }
    const float inv = 1.0f / red[0];
    __syncthreads();

    // ---- phase 2: probs @ V, TDM double-buffered ----
    if (wave == 0) tdm_load_2d(vbase, lds_buf0, g1);

    const int d = tid & 63;          // output dim
    const int g = tid >> 6;          // row group 0..3
    float acc = 0.0f;
    for (int c = 0; c < NCH; ++c) {
        if (wave == 0) {
            if (c + 1 < NCH) {
                tdm_load_2d(vbase + (unsigned long long)(c + 1) * chunk_bytes,
                            ((c + 1) & 1) ? lds_buf1 : lds_buf0, g1);
                __builtin_amdgcn_s_wait_tensorcnt(1);
            } else {
                __builtin_amdgcn_s_wait_tensorcnt(0);
            }
        }
        __syncthreads();
        const float* __restrict__ vb = &buf[c & 1][0];
#pragma unroll
        for (int r = g; r < CHUNK; r += 4)
            acc += sc[c * CHUNK + r] * vb[r * KROW + d];
        __syncthreads();
    }
    if (g == 0) acc += sc[KVN] * vnew[d];

    red[tid] = acc;
    __syncthreads();
    if (tid < 64) {
        const float r = red[tid] + red[tid + 64] + red[tid + 128] + red[tid + 192];
        out[(size_t)b * Hn + h * DHn + tid] = r * inv;
    }
}

// ---------------------------------------------------------------------------
// out = LayerNorm(x + y) * w + b   (one workgroup per row)
// ---------------------------------------------------------------------------
__global__ void __launch_bounds__(256)
add_ln_kernel(const float* __restrict__ x,
              const float* __restrict__ y,
              const float* __restrict__ w,
              const float* __restrict__ bp,
              float* __restrict__ out)
{
    __shared__ float red[256];
    const int b   = blockIdx.x;
    const int tid = threadIdx.x;
    const float* __restrict__ xr = x + (size_t)b * Hn;
    const float* __restrict__ yr = y + (size_t)b * Hn;

    float s = 0.0f;
    for (int i = tid; i < Hn; i += 256) s += xr[i] + yr[i];
    red[tid] = s;
    __syncthreads();
    for (int s2 = 128; s2 > 0; s2 >>= 1) {
        if (tid < s2) red[tid] += red[tid + s2];
        __syncthreads();
    }
    const float mu = red[0] * (1.0f / Hn);
    __syncthreads();

    float v = 0.0f;
    for (int i = tid; i < Hn; i += 256) {
        const float t = xr[i] + yr[i] - mu;
        v += t * t;
    }
    red[tid] = v;
    __syncthreads();
    for (int s2 = 128; s2 > 0; s2 >>= 1) {
        if (tid < s2) red[tid] += red[tid + s2];
        __syncthreads();
    }
    const float rstd = rsqrtf(red[0] * (1.0f / Hn) + 1e-5f);
    __syncthreads();

    for (int i = tid; i < Hn; i += 256) {
        out[(size_t)b * Hn + i] = (xr[i] + yr[i] - mu) * rstd * w[i] + bp[i];
    }
}

// ---------------------------------------------------------------------------
extern "C" void kernel_launch(void* const* d_in, const int* in_sizes, int n_in,
                              void* d_out, int out_size, void* d_ws, size_t ws_size,
                              hipStream_t stream)
{
    (void)in_sizes; (void)n_in; (void)out_size; (void)ws_size;

    const float* x        = (const float*)d_in[0];
    const float* k_cache  = (const float*)d_in[1];
    const float* v_cache  = (const float*)d_in[2];
    const float* qkv_w    = (const float*)d_in[3];
    const float* qkv_b    = (const float*)d_in[4];
    const float* out_w    = (const float*)d_in[5];
    const float* out_b    = (const float*)d_in[6];
    const float* norm_w1  = (const float*)d_in[7];
    const float* norm_b1  = (const float*)d_in[8];
    const float* norm_w2  = (const float*)d_in[9];
    const float* norm_b2  = (const float*)d_in[10];
    const float* mlp_w1   = (const float*)d_in[11];
    const float* mlp_b1   = (const float*)d_in[12];
    const float* mlp_w2   = (const float*)d_in[13];
    const float* mlp_b2   = (const float*)d_in[14];

    float* out = (float*)d_out;            // [x_out | ik | iv], each B*H
    float* ws  = (float*)d_ws;

    float* qkv  = ws;                       // B * 3H
    float* attn = qkv  + Bn * 3 * Hn;       // B * H
    float* tmp  = attn + Bn * Hn;           // B * H
    float* xn   = tmp  + Bn * Hn;           // B * H
    float* hbuf = xn   + Bn * Hn;           // B * FF

    // 1) fused QKV projection
    wmma_gemm_xWT<<<(Bn / 16) * (3 * Hn / 16), 32, 0, stream>>>(
        x, qkv_w, qkv_b, qkv, Bn, 3 * Hn, Hn, 0);

    // 2) emit ik / iv outputs
    split_qkv_kernel<<<(Bn * Hn) / 256, 256, 0, stream>>>(
        qkv, out + Bn * Hn, out + 2 * Bn * Hn);

    // 3) attention over KV cache + appended row (TDM-streamed)
    attn_kernel<<<Bn * NHn, 256, 0, stream>>>(qkv, k_cache, v_cache, attn);

    // 4) output projection
    wmma_gemm_xWT<<<(Bn / 16) * (Hn / 16), 32, 0, stream>>>(
        attn, out_w, out_b, tmp, Bn, Hn, Hn, 0);

    // 5) residual + LN1
    add_ln_kernel<<<Bn, 256, 0, stream>>>(x, tmp, norm_w1, norm_b1, xn);

    // 6) MLP fc1 + ReLU
    wmma_gemm_xWT<<<(Bn / 16) * (FFn / 16), 32, 0, stream>>>(
        xn, mlp_w1, mlp_b1, hbuf, Bn, FFn, Hn, 1);

    // 7) MLP fc2
    wmma_gemm_xWT<<<(Bn / 16) * (Hn / 16), 32, 0, stream>>>(
        hbuf, mlp_w2, mlp_b2, tmp, Bn, Hn, FFn, 0);

    // 8) residual + LN2 -> final output
    add_ln_kernel<<<Bn, 256, 0, stream>>>(xn, tmp, norm_w2, norm_b2, out);
}